// ZWeightedSum_20323785244859
// MI455X (gfx1250) — compile-verified
//
#include <hip/hip_runtime.h>

#define NL 50

typedef __attribute__((ext_vector_type(2))) float v2f;
typedef __attribute__((ext_vector_type(4))) float v4f;
typedef __attribute__((ext_vector_type(8))) float v8f;

// Edges computed at compile time in double precision, bit-matching numpy/jax:
// low_l = (l or 25+7k)/200 * 0.6 -> f32. Bins are contiguous: high[l] == low[l+1]
// (identical double expression), so a single sorted edge array E[0..50] suffices:
// bin l is the strict open interval (E[l], E[l+1]).
constexpr float LOE(int l) {
    return (float)((((l < 25) ? (double)l : (double)(25 + 7 * (l - 25))) / 200.0) * 0.6);
}
constexpr float HIE(int l) {
    return (float)((((l < 25) ? (double)(l + 1) : (double)(32 + 7 * (l - 25))) / 200.0) * 0.6);
}

#define R5(F, a) F(a), F((a) + 1), F((a) + 2), F((a) + 3), F((a) + 4)
#define R25(F, a) R5(F, a), R5(F, (a) + 5), R5(F, (a) + 10), R5(F, (a) + 15), R5(F, (a) + 20)

__constant__ float c_lo[NL] = { R25(LOE, 0), R25(LOE, 25) };

// ds_swizzle xor pattern within 32 lanes: src_lane = lane ^ mask (and_mask=0x1f, or=0)
template <int IMM>
__device__ __forceinline__ float swz_xor_add(float s) {
    int v = __builtin_amdgcn_ds_swizzle(__float_as_int(s), IMM);
    return s + __int_as_float(v);
}

// Fast bin lookup: analytic candidate index, one ds_load_2addr_b64 fetching the
// packed {edge, weight} pair at j0 and j0+1. Slow path (candidate off by one, or
// z exactly on an edge) is statistically ~never taken per wave.
__device__ __forceinline__ float zw_contrib(float z, float v,
                                            const float2* __restrict__ s_tab) {
    float zi = z * 333.33334f;  // z / (0.6/200)
    int j0 = (zi < 25.0f) ? (int)zi : (25 + (int)((zi - 25.0f) * 0.14285715f));
    j0 = j0 < 0 ? 0 : (j0 > NL ? NL : j0);     // j0 in [0, 50]; 50 = "above range" slot
    float2 t0 = s_tab[j0];                      // {E[j0],  w[j0]}   -- one
    float2 t1 = s_tab[j0 + 1];                  // {E[j0+1],w[j0+1]} -- ds_load_2addr_b64
    float w;
    if (__builtin_expect((z > t0.x) && (z < t1.x), 1)) {
        w = t0.y;                               // strictly inside bin j0 (w=0 if j0==50)
    } else {
        // candidate off by one / z on an edge / z <= 0: scan both neighbors
        w = 0.0f;
        int lm = j0 > 0 ? j0 - 1 : 0;
        float2 a0 = s_tab[lm], a1 = s_tab[lm + 1];
        if (z > a0.x && z < a1.x) w = a0.y;
        int lp = j0 < NL ? j0 + 1 : NL;
        float2 b0 = s_tab[lp], b1 = s_tab[lp + 1];
        if (z > b0.x && z < b1.x) w = b0.y;
    }
    return v * w;
}

__global__ __launch_bounds__(256) void ZWeightedSum_kernel(const float* __restrict__ x,
                                                           const float* __restrict__ weight,
                                                           const float* __restrict__ bias,
                                                           float* __restrict__ out) {
    // s_tab[j] = {E[j], w[j]} for j=0..49; [50] = {0.6, 0} top edge; [51] = sentinel.
    __shared__ float2 s_tab[NL + 2];
    __shared__ float red[256];

    const int t = threadIdx.x;
    if (t < NL)           s_tab[t]      = make_float2(c_lo[t], weight[t]);
    else if (t == NL)     s_tab[NL]     = make_float2(HIE(NL - 1), 0.0f);
    else if (t == NL + 1) s_tab[NL + 1] = make_float2(3.0e38f, 0.0f);
    __syncthreads();

    // Block b streams its 4096 (z,val) pairs = 2048 float4: coalesced B128, non-temporal.
    const v4f* __restrict__ xv =
        reinterpret_cast<const v4f*>(x) + (size_t)blockIdx.x * 2048;

    float acc = 0.0f;
#pragma unroll
    for (int i = 0; i < 8; ++i) {
        v4f p = __builtin_nontemporal_load(&xv[i * 256 + t]);
        acc += zw_contrib(p.x, p.y, s_tab);
        acc += zw_contrib(p.z, p.w, s_tab);
    }

    red[t] = acc;
    __syncthreads();

    // Wave 0 reduces 256 partials with 4 accumulated V_WMMA_F32_16X16X4_F32:
    // A = ones(16x4) -> D[m,n] = sum_k B[k,n] + C[m,n]; with an all-ones A the grand
    // total is invariant to how the 64 partials pack into B. After 4 chunks, D VGPR0
    // lanes 0..15 hold the 16 column sums; 4 ds_swizzle xor-adds finish within 16 lanes.
    if (t < 32) {
        v2f a; a[0] = 1.0f; a[1] = 1.0f;
        v8f c = {};
#pragma unroll
        for (int i = 0; i < 4; ++i) {
            v2f bm;
            bm[0] = red[i * 64 + 2 * t];
            bm[1] = red[i * 64 + 2 * t + 1];
            c = __builtin_amdgcn_wmma_f32_16x16x4_f32(
                /*neg_a=*/false, a, /*neg_b=*/false, bm,
                /*c_mod=*/(short)0, c, /*reuse_a=*/false, /*reuse_b=*/false);
        }
        float s = c[0];
        s = swz_xor_add<(8 << 10) | 0x1f>(s);  // lane ^ 8
        s = swz_xor_add<(4 << 10) | 0x1f>(s);  // lane ^ 4
        s = swz_xor_add<(2 << 10) | 0x1f>(s);  // lane ^ 2
        s = swz_xor_add<(1 << 10) | 0x1f>(s);  // lane ^ 1
        if (t == 0) {
            float bw = 0.0f;
            for (int l = 0; l < NL; ++l) bw = fmaf(bias[l], s_tab[l].y, bw);
            out[blockIdx.x] = s + bw;
        }
    }
}

extern "C" void kernel_launch(void* const* d_in, const int* in_sizes, int n_in,
                              void* d_out, int out_size, void* d_ws, size_t ws_size,
                              hipStream_t stream) {
    const float* x      = (const float*)d_in[0];  // (B, 4096, 2) f32
    const float* weight = (const float*)d_in[1];  // (50, 1) f32
    const float* bias   = (const float*)d_in[2];  // (50,) f32
    float* out          = (float*)d_out;          // (B,) f32

    const int B = in_sizes[0] / (4096 * 2);
    ZWeightedSum_kernel<<<dim3(B), dim3(256), 0, stream>>>(x, weight, bias, out);
}